// NCEAverage_1657857376323
// MI455X (gfx1250) — compile-verified
//
#include <hip/hip_runtime.h>
#include <hip/hip_bf16.h>

// ---------------------------------------------------------------------------
// NCEAverage forward for MI455X (gfx1250, wave32, WMMA).
//   out[b,n]  = exp(dot(x[b], memory_da[n,1:]) / T) / Z1[b]
//   Z1[b]     = sum_n exp(...)
//   mask[b,n] = (memory_da[n,0] == (float)labels[b]) ? 1 : 0
// d_out layout: [ out (B*N f32) | mask (B*N f32) ]
// ---------------------------------------------------------------------------

typedef __attribute__((ext_vector_type(2))) float v2f;
typedef __attribute__((ext_vector_type(8))) float v8f;
typedef __attribute__((__vector_size__(16))) int  v4i;

static constexpr int   BB   = 128;      // batch
static constexpr int   DD   = 128;      // feature dim (K)
static constexpr int   NN   = 100000;   // memory bank size
static constexpr int   ROWW = DD + 1;   // 129 floats per memory_da row
static constexpr float INVT = 1.0f / 0.07f;

static constexpr int NTILE  = 64;       // N columns per block
static constexpr int KSTEPS = DD / 4;   // 32 wmma k-steps

// CDNA5 async global->LDS path (ASYNCcnt-tracked), if the toolchain exposes it.
#if defined(__gfx1250__) && \
    __has_builtin(__builtin_amdgcn_global_load_async_to_lds_b128) && \
    __has_builtin(__builtin_amdgcn_s_wait_asynccnt)
#define NCE_ASYNC_LDS 1
typedef __attribute__((address_space(1))) v4i* gv4i_p;   // global int4*
typedef __attribute__((address_space(3))) v4i* lv4i_p;   // LDS int4*
#else
#define NCE_ASYNC_LDS 0
#endif

// ---------------------------------------------------------------------------
__global__ void nce_zero(float* __restrict__ z) {
    if (threadIdx.x < BB) z[threadIdx.x] = 0.0f;
}

// ---------------------------------------------------------------------------
// Pass 1: WMMA GEMM (f32 16x16x4), exp, store unnormalized out, atomic row sums.
// Block: 256 threads (8 waves). Wave w handles rows [16w, 16w+16).
// Block bx handles columns [bx*64, bx*64+64).
// ---------------------------------------------------------------------------
__global__ void __launch_bounds__(256)
nce_pass1(const float* __restrict__ x,
          const float* __restrict__ mda,
          float* __restrict__ outp,
          float* __restrict__ zsum) {
    __shared__ float tile[NTILE * ROWW];          // 33 KB, contiguous slab of memory_da

    const int tid  = threadIdx.x;
    const int lane = tid & 31;
    const int wave = tid >> 5;                    // 0..7
    const int m0   = wave * 16;
    const int n0   = blockIdx.x * NTILE;

    // ---- stage memory_da[n0 : n0+64, :] into LDS (contiguous -> coalesced) ----
    {
        const int rows  = (NN - n0) < NTILE ? (NN - n0) : NTILE;
        const float* __restrict__ src = mda + (size_t)n0 * ROWW;
#if NCE_ASYNC_LDS
        // rows*129 is always a multiple of 4 floats; slab and LDS base are 16B-aligned.
        const int nvec = rows * ROWW / 4;
        for (int i = tid; i < nvec; i += 256) {
            __builtin_amdgcn_global_load_async_to_lds_b128(
                (gv4i_p)(src + 4 * i),
                (lv4i_p)(tile + 4 * i),
                /*offset=*/0, /*cpol=*/0);
        }
        __builtin_amdgcn_s_wait_asynccnt(0);
#else
        const int total = rows * ROWW;
        for (int i = tid; i < total; i += 256) tile[i] = src[i];
#endif
    }

    // ---- preload A fragments for the whole K range into registers ----
    // A 16x4 f32 layout: lanes 0-15 -> (K0,K1) in VGPR0/1, lanes 16-31 -> (K2,K3).
    const int lhalf = lane & 15;                  // M within stripe / N within tile
    const int khalf = (lane >> 4) * 2;            // 0 or 2
    const float* __restrict__ xrow = x + (size_t)(m0 + lhalf) * DD;
    v2f afrag[KSTEPS];
#pragma unroll
    for (int t = 0; t < KSTEPS; ++t) {
        const int k = 4 * t + khalf;
        afrag[t].x = xrow[k];
        afrag[t].y = xrow[k + 1];
    }

    v8f acc[4] = {v8f{0}, v8f{0}, v8f{0}, v8f{0}};

    __syncthreads();

    // ---- main loop: 32 k-steps x 4 n-tiles of v_wmma_f32_16x16x4_f32 ----
#pragma unroll
    for (int t = 0; t < KSTEPS; ++t) {
        const int k = 4 * t + khalf;
#pragma unroll
        for (int j = 0; j < 4; ++j) {
            // B 4x16 f32: lanes 0-15 hold (K0,K1), lanes 16-31 hold (K2,K3), N = lane&15
            const float* __restrict__ bp = &tile[(j * 16 + lhalf) * ROWW + 1 + k];
            v2f b;
            b.x = bp[0];
            b.y = bp[1];
            acc[j] = __builtin_amdgcn_wmma_f32_16x16x4_f32(
                /*neg_a=*/false, afrag[t],
                /*neg_b=*/false, b,
                /*c_mod=*/(short)0, acc[j],
                /*reuse_a=*/false, /*reuse_b=*/false);
        }
    }

    // ---- epilogue: exp, store, per-row partial sums ----
    // C/D layout: VGPR i, lanes 0-15 -> M=m0+i, lanes 16-31 -> M=m0+8+i, N=n0+j*16+(lane&15)
    const int mofs = (lane >> 4) << 3;            // 0 or 8
    float psum[8];
#pragma unroll
    for (int i = 0; i < 8; ++i) psum[i] = 0.0f;

#pragma unroll
    for (int j = 0; j < 4; ++j) {
        if (n0 + j * 16 < NN) {                   // uniform per tile (NN % 16 == 0)
            const int n = n0 + j * 16 + lhalf;
#pragma unroll
            for (int i = 0; i < 8; ++i) {
                const float e = __expf(acc[j][i] * INVT);
                const int m = m0 + mofs + i;
                outp[(size_t)m * NN + n] = e;
                psum[i] += e;
            }
        }
    }

    // xor-shuffle reduce within each 16-lane half (rows differ between halves)
#pragma unroll
    for (int i = 0; i < 8; ++i) {
        float p = psum[i];
        p += __shfl_xor(p, 1, 32);
        p += __shfl_xor(p, 2, 32);
        p += __shfl_xor(p, 4, 32);
        p += __shfl_xor(p, 8, 32);
        if ((lane & 15) == 0) {
            const int m = m0 + mofs + i;
            atomicAdd(&zsum[m], p);
        }
    }
}

// ---------------------------------------------------------------------------
// Pass 2: normalize out by Z1 and emit label-equality mask.
// One thread per column n; loop over the 128 batch rows (coalesced in n).
// ---------------------------------------------------------------------------
__global__ void __launch_bounds__(256)
nce_pass2(const float* __restrict__ mda,
          const int* __restrict__ labels,
          const float* __restrict__ zsum,
          float* __restrict__ outp,
          float* __restrict__ maskp) {
    __shared__ float invz[BB];
    __shared__ float labf[BB];
    if (threadIdx.x < BB) {
        invz[threadIdx.x] = 1.0f / zsum[threadIdx.x];
        labf[threadIdx.x] = (float)labels[threadIdx.x];
    }
    __syncthreads();

    const int n = blockIdx.x * blockDim.x + threadIdx.x;
    if (n >= NN) return;

    const float lbl = mda[(size_t)n * ROWW];      // bank label of column n
#pragma unroll 4
    for (int b = 0; b < BB; ++b) {
        const size_t o = (size_t)b * NN + n;
        outp[o] *= invz[b];
        maskp[o] = (lbl == labf[b]) ? 1.0f : 0.0f;
    }
}

// ---------------------------------------------------------------------------
extern "C" void kernel_launch(void* const* d_in, const int* in_sizes, int n_in,
                              void* d_out, int out_size, void* d_ws, size_t ws_size,
                              hipStream_t stream) {
    const float* x      = (const float*)d_in[0];   // (B, D)
    // d_in[1] = y (unused by reference outputs)
    const int*   labels = (const int*)d_in[2];     // (B,)
    const float* mda    = (const float*)d_in[3];   // (N, D+1)
    // d_in[4] = memory (unused by reference outputs)

    float* outp  = (float*)d_out;                  // (B, N)
    float* maskp = outp + (size_t)BB * NN;         // (B, N)
    float* zsum  = (float*)d_ws;                   // (B,) row sums

    nce_zero<<<1, 128, 0, stream>>>(zsum);

    const int nblk1 = (NN + NTILE - 1) / NTILE;    // 1563
    nce_pass1<<<nblk1, 256, 0, stream>>>(x, mda, outp, zsum);

    const int nblk2 = (NN + 255) / 256;            // 391
    nce_pass2<<<nblk2, 256, 0, stream>>>(mda, labels, zsum, outp, maskp);
}